// LearnedSAGPooling_12927851561634
// MI455X (gfx1250) — compile-verified
//
#include <hip/hip_runtime.h>

#define N_NODES 10000
#define IN_CH   128
#define N_EDGES 160000
#define BATCH   8
#define K_TOP   5000

typedef __attribute__((ext_vector_type(2))) float v2f;
typedef __attribute__((ext_vector_type(8))) float v8f;

// ---------------------------------------------------------------------------
// h = x @ W_gcn + b_gcn  via V_WMMA_F32_16X16X4_F32 (fp32 WMMA, fp32 acc).
// One block = 8 waves; wave w computes the 16x16 tile at cols [16w,16w+16).
// Block handles 16 rows x 128 cols. Grid = BATCH * (N/16) = 5000 blocks.
// Writes result to both h[] and agg[] (agg initialized to h before scatter).
// ---------------------------------------------------------------------------
__global__ void gcn_gemm_wmma(const float* __restrict__ x,
                              const float* __restrict__ W,
                              const float* __restrict__ bias,
                              float* __restrict__ h,
                              float* __restrict__ agg) {
  const int lane = threadIdx.x & 31;
  const int wave = threadIdx.x >> 5;            // 0..7 -> column tile
  const int tiles_per_batch = N_NODES / 16;     // 625
  const int b    = blockIdx.x / tiles_per_batch;
  const int row0 = (blockIdx.x % tiles_per_batch) * 16;
  const int col0 = wave * 16;

  const int m     = lane & 15;                  // row within A tile / col within B tile
  const int khalf = (lane < 16) ? 0 : 2;        // A: lanes 0-15 hold K=0,1; 16-31 hold K=2,3
  const float* xrow = x + ((size_t)b * N_NODES + row0 + m) * IN_CH;

  v8f acc = {};
  #pragma unroll
  for (int k0 = 0; k0 < IN_CH; k0 += 4) {
    v2f a, bm;
    a.x  = xrow[k0 + khalf];
    a.y  = xrow[k0 + khalf + 1];
    bm.x = W[(k0 + khalf) * IN_CH + col0 + m];
    bm.y = W[(k0 + khalf + 1) * IN_CH + col0 + m];
    acc = __builtin_amdgcn_wmma_f32_16x16x4_f32(
        /*neg_a=*/false, a, /*neg_b=*/false, bm,
        /*c_mod=*/(short)0, acc, /*reuse_a=*/false, /*reuse_b=*/false);
  }

  // D layout: vgpr r -> row = r + (lane<16 ? 0 : 8), col = lane%16
  const float bv    = bias[col0 + m];
  const int   rbase = (lane < 16) ? 0 : 8;
  #pragma unroll
  for (int r = 0; r < 8; ++r) {
    const size_t off = ((size_t)b * N_NODES + row0 + rbase + r) * IN_CH + col0 + m;
    const float  v   = acc[r] + bv;
    h[off]   = v;
    agg[off] = v;
  }
}

// ---------------------------------------------------------------------------
// agg[tgt] += h[src] + edge_emb[etype]   (atomic scatter-add, 4 channels/thread)
// ---------------------------------------------------------------------------
__global__ void edge_scatter(const float* __restrict__ h,
                             const int* __restrict__ edge_index,
                             const int* __restrict__ edge_type,
                             const float* __restrict__ edge_emb,
                             float* __restrict__ agg) {
  const long long total = (long long)BATCH * N_EDGES * (IN_CH / 4);
  const long long idx = (long long)blockIdx.x * blockDim.x + threadIdx.x;
  if (idx >= total) return;
  const int c4  = (int)(idx & 31);              // IN_CH/4 == 32
  const long long rem = idx >> 5;
  const int e   = (int)(rem % N_EDGES);
  const int b   = (int)(rem / N_EDGES);
  const int src = edge_index[((size_t)b * 2 + 0) * N_EDGES + e];
  const int tgt = edge_index[((size_t)b * 2 + 1) * N_EDGES + e];
  const int et  = edge_type[(size_t)b * N_EDGES + e];
  const float* hrow = h        + ((size_t)b * N_NODES + src) * IN_CH + c4 * 4;
  const float* erow = edge_emb + (size_t)et * IN_CH + c4 * 4;
  float*       arow = agg      + ((size_t)b * N_NODES + tgt) * IN_CH + c4 * 4;
  #pragma unroll
  for (int i = 0; i < 4; ++i) atomicAdd(&arow[i], hrow[i] + erow[i]);
}

// ---------------------------------------------------------------------------
// scores[n] = relu(agg[n]) . W_score + b_score  (one wave32 per node)
// ---------------------------------------------------------------------------
__global__ void node_score(const float* __restrict__ agg,
                           const float* __restrict__ W_score,
                           const float* __restrict__ b_score,
                           float* __restrict__ scores) {
  const int lane = threadIdx.x & 31;
  const int wv   = (int)((blockIdx.x * (size_t)blockDim.x + threadIdx.x) >> 5);
  if (wv >= BATCH * N_NODES) return;
  const float* row = agg + (size_t)wv * IN_CH;
  float s = 0.f;
  #pragma unroll
  for (int i = 0; i < 4; ++i) {
    const int c = lane * 4 + i;
    const float v = row[c];
    s += (v > 0.f ? v : 0.f) * W_score[c];
  }
  #pragma unroll
  for (int off = 16; off > 0; off >>= 1) s += __shfl_xor(s, off, 32);
  if (lane == 0) scores[wv] = s + b_score[0];
}

// ---------------------------------------------------------------------------
// Exact ordered top-K via ranking: rank(i) = #{ j : s_j > s_i  or
// (s_j == s_i and j < i) }. Matches jax.lax.top_k ordering (desc, stable).
// One wave per node; node i goes to output slot rank(i) if rank < K.
// ---------------------------------------------------------------------------
__global__ void topk_rank(const float* __restrict__ scores,
                          int* __restrict__ topk) {
  const int lane = threadIdx.x & 31;
  const int wv   = (int)((blockIdx.x * (size_t)blockDim.x + threadIdx.x) >> 5);
  if (wv >= BATCH * N_NODES) return;
  const int b = wv / N_NODES;
  const int i = wv % N_NODES;
  const float* sb = scores + (size_t)b * N_NODES;
  const float si = sb[i];
  int cnt = 0;
  for (int j = lane; j < N_NODES; j += 32) {
    const float sj = sb[j];
    cnt += (sj > si) || (sj == si && j < i);
  }
  #pragma unroll
  for (int off = 16; off > 0; off >>= 1) cnt += __shfl_xor(cnt, off, 32);
  if (lane == 0 && cnt < K_TOP) topk[(size_t)b * K_TOP + cnt] = i;
}

// ---------------------------------------------------------------------------
// x_top[b,r,:] = x[b, topk[b,r], :]  (written to d_out) and
// attn[b,r]    = x_top[b,r,:] . W_attn + b_attn
// ---------------------------------------------------------------------------
__global__ void gather_attn(const float* __restrict__ x,
                            const int* __restrict__ topk,
                            const float* __restrict__ W_attn,
                            const float* __restrict__ b_attn,
                            float* __restrict__ x_top_out,
                            float* __restrict__ attn) {
  __shared__ float red[IN_CH];
  const int b = blockIdx.x / K_TOP;
  const int r = blockIdx.x % K_TOP;
  const int c = threadIdx.x;                    // 128 threads
  const int idx = topk[(size_t)b * K_TOP + r];
  const float v = x[((size_t)b * N_NODES + idx) * IN_CH + c];
  x_top_out[((size_t)b * K_TOP + r) * IN_CH + c] = v;
  red[c] = v * W_attn[c];
  __syncthreads();
  for (int s = IN_CH / 2; s > 0; s >>= 1) {
    if (c < s) red[c] += red[c + s];
    __syncthreads();
  }
  if (c == 0) attn[(size_t)b * K_TOP + r] = red[0] + b_attn[0];
}

// ---------------------------------------------------------------------------
// Per-batch softmax statistics (max, sum of exp) over K attention scores.
// ---------------------------------------------------------------------------
__global__ void softmax_stats(const float* __restrict__ attn,
                              float* __restrict__ stats) {
  __shared__ float red[256];
  const int b = blockIdx.x;
  const int t = threadIdx.x;
  const float* a = attn + (size_t)b * K_TOP;
  float mx = -3.402823466e38f;
  for (int r = t; r < K_TOP; r += 256) mx = fmaxf(mx, a[r]);
  red[t] = mx; __syncthreads();
  for (int s = 128; s > 0; s >>= 1) {
    if (t < s) red[t] = fmaxf(red[t], red[t + s]);
    __syncthreads();
  }
  mx = red[0]; __syncthreads();
  float sum = 0.f;
  for (int r = t; r < K_TOP; r += 256) sum += __expf(a[r] - mx);
  red[t] = sum; __syncthreads();
  for (int s = 128; s > 0; s >>= 1) {
    if (t < s) red[t] += red[t + s];
    __syncthreads();
  }
  if (t == 0) { stats[b * 2] = mx; stats[b * 2 + 1] = red[0]; }
}

// ---------------------------------------------------------------------------
// pooled[b,c] = sum_r softmax(attn)[r] * x_top[b,r,c]
// ---------------------------------------------------------------------------
__global__ void pooled_kernel(const float* __restrict__ x_top,
                              const float* __restrict__ attn,
                              const float* __restrict__ stats,
                              float* __restrict__ pooled) {
  const int b = blockIdx.x;
  const int c = threadIdx.x;                    // 128 threads
  const float mx  = stats[b * 2];
  const float inv = 1.0f / stats[b * 2 + 1];
  const float* a  = attn + (size_t)b * K_TOP;
  const float* xt = x_top + (size_t)b * K_TOP * IN_CH + c;
  float acc = 0.f;
  for (int r = 0; r < K_TOP; ++r)
    acc += __expf(a[r] - mx) * inv * xt[(size_t)r * IN_CH];
  pooled[(size_t)b * IN_CH + c] = acc;
}

extern "C" void kernel_launch(void* const* d_in, const int* in_sizes, int n_in,
                              void* d_out, int out_size, void* d_ws, size_t ws_size,
                              hipStream_t stream) {
  const float* x       = (const float*)d_in[0];
  const int*   e_idx   = (const int*)  d_in[1];
  const int*   e_type  = (const int*)  d_in[2];
  const float* e_emb   = (const float*)d_in[3];
  const float* W_gcn   = (const float*)d_in[4];
  const float* b_gcn   = (const float*)d_in[5];
  const float* W_score = (const float*)d_in[6];
  const float* b_score = (const float*)d_in[7];
  const float* W_attn  = (const float*)d_in[8];
  const float* b_attn  = (const float*)d_in[9];

  // Output layout (tuple flattened): pooled (B,1,C) then x_top (B,K,C)
  float* pooled = (float*)d_out;
  float* x_top  = (float*)d_out + (size_t)BATCH * IN_CH;

  // Workspace layout
  float* h      = (float*)d_ws;
  float* agg    = h      + (size_t)BATCH * N_NODES * IN_CH;
  float* scores = agg    + (size_t)BATCH * N_NODES * IN_CH;
  int*   topk   = (int*)(scores + (size_t)BATCH * N_NODES);
  float* attn   = (float*)(topk + (size_t)BATCH * K_TOP);
  float* stats  = attn   + (size_t)BATCH * K_TOP;

  // 1) GCN linear layer via fp32 WMMA; also initializes agg = h
  gcn_gemm_wmma<<<BATCH * (N_NODES / 16), 256, 0, stream>>>(x, W_gcn, b_gcn, h, agg);

  // 2) Edge message scatter-add
  {
    const long long total = (long long)BATCH * N_EDGES * (IN_CH / 4);
    const int blocks = (int)((total + 255) / 256);
    edge_scatter<<<blocks, 256, 0, stream>>>(h, e_idx, e_type, e_emb, agg);
  }

  // 3) relu + score projection (one wave per node; 8 waves/block)
  node_score<<<(BATCH * N_NODES) / 8, 256, 0, stream>>>(agg, W_score, b_score, scores);

  // 4) ordered top-K via exact ranking
  topk_rank<<<(BATCH * N_NODES) / 8, 256, 0, stream>>>(scores, topk);

  // 5) gather x_top into output + attention scores
  gather_attn<<<BATCH * K_TOP, IN_CH, 0, stream>>>(x, topk, W_attn, b_attn, x_top, attn);

  // 6) softmax stats per batch
  softmax_stats<<<BATCH, 256, 0, stream>>>(attn, stats);

  // 7) weighted pooling into output
  pooled_kernel<<<BATCH, IN_CH, 0, stream>>>(x_top, attn, stats, pooled);
}